// CRF_22582938042591
// MI455X (gfx1250) — compile-verified
//
#include <hip/hip_runtime.h>
#include <stdint.h>

// Problem dims (fixed by the reference)
#define S_LEN 512
#define B_SZ  256
#define T_TAGS 128
#define TPAD  129   // padded row stride (floats) for transposed transitions: bank (j*129+i)%64 = (j+i)%64 -> conflict-free

// LDS layout (bytes):
//   transT : 128 * 129 * 4 = 66048   (transitions transposed, row j holds trans[:,j])
//   score  : 2 * 128 * 4   = 1024    (per-batch-lane Viterbi scores)
//   embuf  : 2 * 256 * 4   = 2048    (double-buffered emissions rows for 2 batches)
//   hist   : 2 * 511 * 128 = 130816  (u8 argmax history, kept on-chip for backtrack)
#define OFF_SCORE 66048
#define OFF_EM    (66048 + 1024)
#define OFF_HIST  (66048 + 1024 + 2048)
#define SMEM_BYTES (66048 + 1024 + 2048 + 130816)   // 199936 < 320KB WGP LDS

typedef __attribute__((address_space(1))) int gint_t;   // global (AS1) int
typedef __attribute__((address_space(3))) int lint_t;   // LDS (AS3) int

// ---- CDNA5 async global->LDS copy (per-lane scattered LDS destination) ----
__device__ __forceinline__ void async_copy_b32(const void* gptr, void* lds_generic) {
#if __has_builtin(__builtin_amdgcn_global_load_async_to_lds_b32)
  __builtin_amdgcn_global_load_async_to_lds_b32(
      (gint_t*)(uintptr_t)gptr,
      (lint_t*)(uint32_t)(uintptr_t)lds_generic,
      0, 0);
#else
  uint32_t lds_off = (uint32_t)(uintptr_t)lds_generic;   // low 32 bits of generic LDS addr = LDS byte offset
  uint64_t gaddr   = (uint64_t)(uintptr_t)gptr;
  asm volatile("global_load_async_to_lds_b32 %0, %1, off"
               :: "v"(lds_off), "v"(gaddr) : "memory");
#endif
}

template <int N>
__device__ __forceinline__ void wait_asynccnt() {
#if __has_builtin(__builtin_amdgcn_s_wait_asynccnt)
  __builtin_amdgcn_s_wait_asynccnt(N);
#else
  asm volatile("s_wait_asynccnt %0" :: "i"(N) : "memory");
#endif
}

// One block = 256 threads = 2 batch lanes (threads 0..127 -> b0, 128..255 -> b0+1).
// Full Viterbi (forward max-plus recursion + on-chip backtrack) per batch lane.
extern "C" __global__ __launch_bounds__(256)
void crf_viterbi_kernel(const float* __restrict__ em,      // [S,B,T]
                        const int*   __restrict__ mask,    // [S,B]
                        const float* __restrict__ start_t, // [T]
                        const float* __restrict__ end_t,   // [T]
                        const float* __restrict__ trans,   // [T,T]
                        int*   __restrict__ tags_out,      // [B,S]
                        float* __restrict__ score_out)     // [B]
{
  extern __shared__ char smem[];
  float*   transT = (float*)smem;
  float*   score  = (float*)(smem + OFF_SCORE);
  float*   embuf  = (float*)(smem + OFF_EM);
  uint8_t* hist   = (uint8_t*)(smem + OFF_HIST);

  const int tid = threadIdx.x;
  const int bl  = tid >> 7;       // local batch lane 0/1
  const int j   = tid & 127;      // tag owned by this thread
  const int b0  = blockIdx.x * 2;
  const int bg  = b0 + bl;        // global batch index

  // ---- async transposed load of transitions: transT[j][i] = trans[i][j] ----
  for (int e = 0; e < 64; ++e) {
    int linear = e * 256 + tid;        // linear = i*128 + jj
    int i  = linear >> 7;
    int jj = linear & 127;
    async_copy_b32(trans + linear, &transT[jj * TPAD + i]);
  }
  // ---- prefetch emissions row s=0 for both batch lanes (256 contiguous floats) ----
  async_copy_b32(em + (size_t)b0 * T_TAGS + tid, &embuf[tid]);
  wait_asynccnt<0>();
  __syncthreads();

  float* sb = score + bl * T_TAGS;
  // score0 = start_transitions + emissions[0]
  sb[j] = start_t[j] + embuf[bl * T_TAGS + j];
  __syncthreads();

  // prefetch s=1 into buffer 1
  async_copy_b32(em + ((size_t)1 * B_SZ + b0) * T_TAGS + tid, &embuf[256 + tid]);

  const float* trow = transT + j * TPAD;
  uint8_t* hrow = hist + (size_t)bl * (511 * 128);

  for (int s = 1; s < S_LEN; ++s) {
    // prefetch s+1 (double buffer), then wait for s's data
    if (s + 1 < S_LEN) {
      async_copy_b32(em + ((size_t)(s + 1) * B_SZ + b0) * T_TAGS + tid,
                     &embuf[((s + 1) & 1) * 256 + tid]);
      wait_asynccnt<1>();
    } else {
      wait_asynccnt<0>();
    }
    __syncthreads();

    const float* eb = embuf + (s & 1) * 256;
    float emv = eb[bl * 128 + j];
    int   mk  = mask[s * B_SZ + bg];

    // next[j] = max_i (score[i] + transT[j][i]); track first-argmax
    const float4* sv = (const float4*)sb;   // 16B-aligned broadcast reads
    float best = -3.402823466e38f;
    int   bi   = 0;
#pragma unroll 8
    for (int i4 = 0; i4 < 32; ++i4) {
      float4 sc = sv[i4];
      int ib = i4 * 4;
      float v0 = sc.x + trow[ib + 0];
      float v1 = sc.y + trow[ib + 1];
      float v2 = sc.z + trow[ib + 2];
      float v3 = sc.w + trow[ib + 3];
      if (v0 > best) { best = v0; bi = ib + 0; }
      if (v1 > best) { best = v1; bi = ib + 1; }
      if (v2 > best) { best = v2; bi = ib + 2; }
      if (v3 > best) { best = v3; bi = ib + 3; }
    }
    float ns   = best + emv;
    float oldv = sb[j];
    hrow[(s - 1) * 128 + j] = (uint8_t)bi;   // history for step s

    __syncthreads();                         // all reads of score done
    sb[j] = mk ? ns : oldv;                  // masked update
  }
  __syncthreads();

  // final_score = score + end_transitions
  float fs = sb[j] + end_t[j];
  sb[j] = fs;
  __syncthreads();

  // one thread per batch lane: argmax + on-chip backtrack into staged tags
  int* tagstage = (int*)transT;              // reuse transT region: 2*512 ints
  if (j == 0) {
    float bs = sb[0]; int bt = 0;
    for (int t = 1; t < T_TAGS; ++t) {
      float v = sb[t];
      if (v > bs) { bs = v; bt = t; }
    }
    score_out[bg] = bs;
    int tag = bt;
    int* ts = tagstage + bl * S_LEN;
    ts[S_LEN - 1] = tag;
    for (int s = S_LEN - 2; s >= 0; --s) {
      int prev = hrow[s * 128 + tag];
      if (mask[(s + 1) * B_SZ + bg] != 0) tag = prev;
      ts[s] = tag;
    }
  }
  __syncthreads();

  // coalesced write of staged tags: tags_out[b][s]
  for (int idx = tid; idx < 2 * S_LEN; idx += 256) {
    int bl2 = idx >> 9;          // /512
    int s2  = idx & (S_LEN - 1);
    tags_out[(size_t)(b0 + bl2) * S_LEN + s2] = tagstage[idx];
  }
}

extern "C" void kernel_launch(void* const* d_in, const int* in_sizes, int n_in,
                              void* d_out, int out_size, void* d_ws, size_t ws_size,
                              hipStream_t stream) {
  (void)in_sizes; (void)n_in; (void)out_size; (void)d_ws; (void)ws_size;
  const float* em     = (const float*)d_in[0];   // [S,B,T] f32
  const int*   mask   = (const int*)d_in[1];     // [S,B] i32
  const float* startt = (const float*)d_in[2];   // [T]
  const float* endt   = (const float*)d_in[3];   // [T]
  const float* trans  = (const float*)d_in[4];   // [T,T]

  // outputs concatenated: tags [B,S] int32, then best_score [B] float32
  int*   tags  = (int*)d_out;
  float* score = (float*)((char*)d_out + (size_t)B_SZ * S_LEN * sizeof(int));

  dim3 grid(B_SZ / 2);   // 128 blocks, 2 batch lanes each
  dim3 block(256);
  crf_viterbi_kernel<<<grid, block, SMEM_BYTES, stream>>>(
      em, mask, startt, endt, trans, tags, score);
}